// RetinaNet_Criteria_18451179504190
// MI455X (gfx1250) — compile-verified
//
#include <hip/hip_runtime.h>
#include <math.h>

// RetinaNet criteria (CrowdDet) fused loss for MI455X / gfx1250 (wave32).
// Stage 1: zero 3 accumulators. Stage 2: fused assign+focal+smoothL1, block
// reduction via V_WMMA_F32_16X16X4_F32 + ballot, global f32 atomics.
// Stage 3: normalizer + final divide.

typedef float v2f __attribute__((ext_vector_type(2)));
typedef float v8f __attribute__((ext_vector_type(8)));

#define POS_THRESH    0.5f
#define NEG_THRESH    0.4f
#define FOCAL_ALPHA   0.25f
#define FOCAL_GAMMA_IS_2 1
#define SL1_BETA      0.1f
#define LOSS_NORM0    100.0f
#define NORM_MOM      0.9f
#define BLOCK         256

__global__ void ws_init_kernel(float* __restrict__ ws) {
    if (threadIdx.x < 3) ws[threadIdx.x] = 0.0f;
}

__global__ __launch_bounds__(BLOCK)
void retina_loss_kernel(const float* __restrict__ pred_cls,
                        const float* __restrict__ pred_reg,
                        const float* __restrict__ anchors,
                        const float* __restrict__ gt_boxes,
                        const float* __restrict__ im_info,
                        int A, int G, float* __restrict__ ws) {
    extern __shared__ float4 smem4[];
    float4* s_box  = smem4;                       // (x0, y0, x1+1, y1+1)
    float2* s_meta = (float2*)(smem4 + G);        // (area_legacy, class)
    float*  s_red  = (float*)(s_meta + G);        // 24 floats used

    const int b   = blockIdx.y;
    const int tid = threadIdx.x;

    // ---- stage GT boxes into LDS (precompute +1-shifted corners & areas) ----
    for (int g = tid; g < G; g += BLOCK) {
        const float* p = gt_boxes + ((size_t)b * G + g) * 5;
        const float x0 = p[0], y0 = p[1];
        const float x1p = p[2] + 1.0f, y1p = p[3] + 1.0f;   // legacy +1 convention
        s_box[g]  = make_float4(x0, y0, x1p, y1p);
        s_meta[g] = make_float2((x1p - x0) * (y1p - y0), p[4]);
    }
    __syncthreads();

    int nv = (int)im_info[b * 6 + 5];
    nv = nv < 0 ? 0 : (nv > G ? G : nv);

    const int  a    = blockIdx.x * BLOCK + tid;
    const bool live = (a < A);
    const int  ac   = live ? a : (A - 1);   // clamp: keep full EXEC for WMMA

    const float ax0  = anchors[ac * 4 + 0];
    const float ay0  = anchors[ac * 4 + 1];
    const float ax1p = anchors[ac * 4 + 2] + 1.0f;
    const float ay1p = anchors[ac * 4 + 3] + 1.0f;
    const float aw = ax1p - ax0, ah = ay1p - ay0;
    const float area_a = aw * ah;

    // ---- division-free argmax over valid GTs (first-max-wins like jnp.argmax) ----
    float bn = -2.0f, bd = 1.0f;    // best IoU as fraction bn/bd, bd > 0
    int   arg = 0;
    #pragma unroll 4
    for (int g = 0; g < nv; ++g) {
        const float4 gb = s_box[g];
        const float2 gm = s_meta[g];
        const float iw    = fminf(ax1p, gb.z) - fmaxf(ax0, gb.x);
        const float ih    = fminf(ay1p, gb.w) - fmaxf(ay0, gb.y);
        const float inter = fmaxf(iw, 0.0f) * fmaxf(ih, 0.0f);
        const float den   = area_a + gm.x - inter;          // > 0 always
        const bool  upd   = inter * bd > bn * den;          // strict: keep first max
        bn  = upd ? inter : bn;
        bd  = upd ? den   : bd;
        arg = upd ? g     : arg;
    }
    const float max_ov = bn / bd;

    // ---- label + bbox-transform target for matched GT ----
    const float4 gb = s_box[arg];
    const float2 gm = s_meta[arg];
    const float gw = gb.z - gb.x, gh = gb.w - gb.y;         // w = x1-x0+1
    const float t0 = (gb.x + 0.5f * gw - (ax0 + 0.5f * aw)) / aw;
    const float t1 = (gb.y + 0.5f * gh - (ay0 + 0.5f * ah)) / ah;
    const float t2 = logf(gw / aw);
    const float t3 = logf(gh / ah);

    const float label = (max_ov >= POS_THRESH) ? gm.y
                      : ((max_ov >= NEG_THRESH) ? -1.0f : 0.0f);
    const float validf = (label >= 0.0f) ? 1.0f : 0.0f;
    float       fg     = (label >  0.0f) ? 1.0f : 0.0f;
    const float pos    = (label == 1.0f) ? 1.0f : 0.0f;     // class_range == [1]

    // ---- focal loss (stable log-sigmoid; gamma = 2) ----
    const float x    = pred_cls[(size_t)b * A + ac];
    const float sp   = log1pf(expf(-fabsf(x)));             // softplus(-|x|)
    const float lsp  = -(fmaxf(-x, 0.0f) + sp);             // log sigmoid(x)
    const float lsn  = -(fmaxf( x, 0.0f) + sp);             // log sigmoid(-x)
    const float p    = expf(lsp);
    const float omp  = 1.0f - p;
    const float focal = -(FOCAL_ALPHA * pos * omp * omp * lsp
                        + (1.0f - FOCAL_ALPHA) * (1.0f - pos) * p * p * lsn);
    float cls_l = focal * validf;

    // ---- smooth L1 on fg anchors ----
    const float* rg = pred_reg + ((size_t)b * A + ac) * 4;
    const float tt[4] = {t0, t1, t2, t3};
    float reg_l = 0.0f;
    #pragma unroll
    for (int i = 0; i < 4; ++i) {
        const float d = fabsf(rg[i] - tt[i]);
        reg_l += (d < SL1_BETA) ? (0.5f / SL1_BETA) * d * d : d - SL1_BETA;
    }
    reg_l *= fg;

    if (!live) { cls_l = 0.0f; reg_l = 0.0f; fg = 0.0f; }   // reconverges before WMMA

    // ---- wave reduction via V_WMMA_F32_16X16X4_F32 ----
    // A: vgpr0 = cls partial, vgpr1 = reg partial (16x4 f32, K striped lane-halves).
    // B: selector so D[:,0] = sum of cls over K, D[:,1] = sum of reg over K.
    const int lane = tid & 31;
    const int wid  = tid >> 5;
    v2f am; am[0] = cls_l; am[1] = reg_l;
    v2f bm; bm[0] = ((lane & 15) == 0) ? 1.0f : 0.0f;       // rows K=0,2 -> col 0
            bm[1] = ((lane & 15) == 1) ? 1.0f : 0.0f;       // rows K=1,3 -> col 1
    v8f cm = {};
    v8f dm = __builtin_amdgcn_wmma_f32_16x16x4_f32(
        false, am, false, bm, (short)0, cm, false, false);
    float s = dm[0] + dm[1] + dm[2] + dm[3] + dm[4] + dm[5] + dm[6] + dm[7];
    s += __shfl_xor(s, 16);   // lanes 0/16: wave cls sum; lanes 1/17: wave reg sum

    const unsigned long long bal = __ballot(fg > 0.0f);     // wave32: low 32 bits
    const float fgw = (float)__popcll(bal);

    if (lane == 0) { s_red[wid] = s; s_red[16 + wid] = fgw; }
    if (lane == 1) { s_red[8 + wid] = s; }
    __syncthreads();

    if (tid == 0) {
        float c = 0.0f, r = 0.0f, f = 0.0f;
        #pragma unroll
        for (int w = 0; w < BLOCK / 32; ++w) {
            c += s_red[w]; r += s_red[8 + w]; f += s_red[16 + w];
        }
        atomicAdd(&ws[0], c);
        atomicAdd(&ws[1], r);
        atomicAdd(&ws[2], f);
    }
}

__global__ void finalize_kernel(const float* __restrict__ ws, float* __restrict__ out) {
    if (threadIdx.x == 0) {
        const float num_pos    = fmaxf(ws[2], 1.0f);
        const float normalizer = NORM_MOM * LOSS_NORM0 + (1.0f - NORM_MOM) * num_pos;
        out[0] = ws[0] / normalizer;
        out[1] = ws[1] / normalizer;
    }
}

extern "C" void kernel_launch(void* const* d_in, const int* in_sizes, int n_in,
                              void* d_out, int out_size, void* d_ws, size_t ws_size,
                              hipStream_t stream) {
    const float* pred_cls = (const float*)d_in[0];
    const float* pred_reg = (const float*)d_in[1];
    const float* anchors  = (const float*)d_in[2];
    const float* gt_boxes = (const float*)d_in[3];
    const float* im_info  = (const float*)d_in[4];
    float* out = (float*)d_out;
    float* ws  = (float*)d_ws;

    const int A = in_sizes[2] / 4;            // anchors [A,4]
    const int B = in_sizes[4] / 6;            // im_info [B,6]
    const int G = in_sizes[3] / (B * 5);      // gt_boxes [B,G,5]

    hipLaunchKernelGGL(ws_init_kernel, dim3(1), dim3(32), 0, stream, ws);

    dim3 grid((A + BLOCK - 1) / BLOCK, B);
    size_t shmem = (size_t)G * (sizeof(float4) + sizeof(float2)) + 32 * sizeof(float);
    hipLaunchKernelGGL(retina_loss_kernel, grid, dim3(BLOCK), shmem, stream,
                       pred_cls, pred_reg, anchors, gt_boxes, im_info, A, G, ws);

    hipLaunchKernelGGL(finalize_kernel, dim3(1), dim3(1), 0, stream, ws, out);
}